// msdecomp_38671885533223
// MI455X (gfx1250) — compile-verified
//
#include <hip/hip_runtime.h>

typedef __attribute__((ext_vector_type(2))) float v2f;
typedef __attribute__((ext_vector_type(8))) float v8f;

#define BATCH 64
#define LDIM 96
#define CDIM 2048
#define TILE 128
#define TCOLS (TILE + 2)   // 130 columns incl. circular halo
#define KTOT 288           // im2col K = 3*96
#define KSTEPS (KTOT / 4)  // 72 WMMA K-steps of 4

// Pre-pack conv weights into im2col layout, K-pair interleaved:
// wt2 float index = (kp*96 + m)*2 + j  holds  w_im2col[m][2*kp + j],
// with k = dt*96 + ci  ->  conv_w[(m*96 + ci)*3 + dt]
__global__ void build_wt_kernel(const float* __restrict__ conv_w,
                                float* __restrict__ wt2) {
  int idx = blockIdx.x * blockDim.x + threadIdx.x;
  if (idx >= KTOT * LDIM) return;          // 27648 floats total
  int j  = idx & 1;
  int m  = (idx >> 1) % LDIM;
  int kp = idx / (2 * LDIM);
  int kk = 2 * kp + j;
  int ci = kk % LDIM;
  int dt = kk / LDIM;
  wt2[idx] = conv_w[(m * LDIM + ci) * 3 + dt];
}

__global__ __launch_bounds__(192)
void msdecomp_conv_kernel(const float* __restrict__ x,
                          const float* __restrict__ wt2,
                          const float* __restrict__ gate_w,
                          const float* __restrict__ gate_b,
                          float* __restrict__ out) {
  __shared__ float xs[TCOLS * LDIM];   // x tile, [col][l]
  __shared__ float rs[TCOLS * LDIM];   // res tile, [col][l]
  __shared__ float ms[TCOLS * LDIM];   // mean tile, [col][l]

  const int tid = threadIdx.x;
  const int b   = blockIdx.y;
  const int c0  = blockIdx.x * TILE;
  const float* xb = x + (size_t)b * LDIM * CDIM;

  // ---- load x tile (transposed into LDS), circular halo in C ----
  for (int i = tid; i < TCOLS * LDIM; i += blockDim.x) {
    int l   = i / TCOLS;
    int col = i % TCOLS;
    int gc  = (c0 + col - 1 + CDIM) & (CDIM - 1);
    xs[col * LDIM + l] = xb[l * CDIM + gc];
  }
  __syncthreads();

  // ---- decomposition: sliding windows along L (replicate pad = clamp) ----
  if (tid < TCOLS) {
    const float gw0 = gate_w[0], gw1 = gate_w[1], gw2 = gate_w[2];
    const float gb0 = gate_b[0], gb1 = gate_b[1], gb2 = gate_b[2];
    const float* xc = &xs[tid * LDIM];
    float* rc = &rs[tid * LDIM];
    float* mc = &ms[tid * LDIM];

    float s5 = 3.f * xc[0] + xc[1] + xc[2];              // k=5,  p=2
    float s13 = 7.f * xc[0];                             // k=13, p=6
    #pragma unroll
    for (int j = 1; j <= 6; ++j) s13 += xc[j];
    float s25 = 13.f * xc[0];                            // k=25, p=12
    #pragma unroll
    for (int j = 1; j <= 12; ++j) s25 += xc[j];

    for (int l = 0; l < LDIM; ++l) {
      float xv  = xc[l];
      float m5  = s5 * 0.2f;
      float m13 = s13 * (1.f / 13.f);
      float m25 = s25 * 0.04f;
      float l0 = fmaf(xv, gw0, gb0);
      float l1 = fmaf(xv, gw1, gb1);
      float l2 = fmaf(xv, gw2, gb2);
      float mx = fmaxf(fmaxf(l0, l1), l2);
      float e0 = __expf(l0 - mx), e1 = __expf(l1 - mx), e2 = __expf(l2 - mx);
      float inv = 1.f / (e0 + e1 + e2);
      float mean = (m5 * e0 + m13 * e1 + m25 * e2) * inv;
      mc[l] = mean;
      rc[l] = xv - mean;
      s5  += xc[min(l + 3, 95)]  - xc[max(l - 2, 0)];
      s13 += xc[min(l + 7, 95)]  - xc[max(l - 6, 0)];
      s25 += xc[min(l + 13, 95)] - xc[max(l - 12, 0)];
    }
  }
  __syncthreads();

  // ---- conv as GEMM on V_WMMA_F32_16X16X4_F32 ----
  // wave = M-tile (6 waves x 16 rows = 96 output channels)
  // per wave: 16 accumulator tiles = {res,mean} x 8 N-tiles of 16 cols
  const int wave = tid >> 5;
  const int lane = tid & 31;
  const int half = (lane >> 4) & 1;   // K lo/hi pair split (k 0-1 vs 2-3 of each step)
  const int ln   = lane & 15;
  const int m0   = wave * 16;

  v8f acc[16];
  const v8f vz = {0.f, 0.f, 0.f, 0.f, 0.f, 0.f, 0.f, 0.f};
  #pragma unroll
  for (int t = 0; t < 16; ++t) acc[t] = vz;

  const v2f* wt = (const v2f*)wt2;
  #pragma unroll 2
  for (int ks = 0; ks < KSTEPS; ++ks) {
    // A fragment: 16x4 slice of im2col weights (L2-resident, coalesced b64)
    v2f A = wt[(2 * ks + half) * LDIM + m0 + ln];
    int dt = ks / 24;                 // (4*ks)/96
    int ci = (4 * ks) % 96 + 2 * half;
    #pragma unroll
    for (int t = 0; t < 16; ++t) {
      const float* src = (t < 8) ? rs : ms;
      int row = (t & 7) * 16 + ln + dt;           // column in C-tile (+halo dt)
      v2f Bf = *(const v2f*)&src[row * LDIM + ci]; // aligned float2 from LDS
      acc[t] = __builtin_amdgcn_wmma_f32_16x16x4_f32(
          false, A, false, Bf, (short)0, acc[t], false, false);
    }
  }

  // ---- store D tiles: out[conv][b][co][c] ----
  #pragma unroll
  for (int t = 0; t < 16; ++t) {
    int conv  = t >> 3;
    int ntile = t & 7;
    float* op = out + (size_t)conv * BATCH * LDIM * CDIM
                    + (size_t)b * LDIM * CDIM;
    int cc = c0 + ntile * 16 + ln;
    #pragma unroll
    for (int r = 0; r < 8; ++r) {
      int rowm = m0 + r + 8 * half;
      op[(size_t)rowm * CDIM + cc] = acc[t][r];
    }
  }
}

extern "C" void kernel_launch(void* const* d_in, const int* in_sizes, int n_in,
                              void* d_out, int out_size, void* d_ws, size_t ws_size,
                              hipStream_t stream) {
  const float* x      = (const float*)d_in[0];
  const float* conv_w = (const float*)d_in[1];
  const float* gate_w = (const float*)d_in[2];
  const float* gate_b = (const float*)d_in[3];
  float* out = (float*)d_out;
  float* wt2 = (float*)d_ws;   // 288*96 floats = 110,592 B of workspace

  build_wt_kernel<<<(KTOT * LDIM + 255) / 256, 256, 0, stream>>>(conv_w, wt2);

  dim3 grid(CDIM / TILE, BATCH);
  msdecomp_conv_kernel<<<grid, 192, 0, stream>>>(x, wt2, gate_w, gate_b, out);
}